// JensenShannonDivergenceWeightedScaled_21517786153524
// MI455X (gfx1250) — compile-verified
//
#include <hip/hip_runtime.h>
#include <hip/hip_bf16.h>
#include <stdint.h>

#define AS1 __attribute__((address_space(1)))
#define AS3 __attribute__((address_space(3)))

typedef int v4i __attribute__((ext_vector_type(4)));

#ifndef __has_builtin
#define __has_builtin(x) 0
#endif

// ---------------------------------------------------------------------------
// CDNA5 async global->LDS copy (ASYNCcnt-tracked). Builtin prototype (from
// hipcc diagnostic): (v4i AS1*, v4i AS3*, Imm i32 offset, Imm i32 cpol).
// ---------------------------------------------------------------------------
__device__ __forceinline__ void async_copy_b128(const void* gsrc, void* ldst) {
#if __has_builtin(__builtin_amdgcn_global_load_async_to_lds_b128)
    __builtin_amdgcn_global_load_async_to_lds_b128((AS1 v4i*)gsrc, (AS3 v4i*)ldst, 0, 0);
#else
    unsigned lds_off = (unsigned)(uintptr_t)(AS3 void*)ldst;
    asm volatile("global_load_async_to_lds_b128 %0, %1, off"
                 :: "v"(lds_off), "v"(gsrc)
                 : "memory");
#endif
}

template <int N>
__device__ __forceinline__ void wait_asynccnt() {
#if __has_builtin(__builtin_amdgcn_s_wait_asynccnt)
    __builtin_amdgcn_s_wait_asynccnt(N);
#else
    if (N == 0) asm volatile("s_wait_asynccnt 0" ::: "memory");
    else if (N == 3) asm volatile("s_wait_asynccnt 3" ::: "memory");
    else asm volatile("s_wait_asynccnt 0" ::: "memory");
#endif
}

// ---------------------------------------------------------------------------
// Problem constants (from reference setup_inputs)
// ---------------------------------------------------------------------------
#define BB    512
#define SS    16384
#define NT    512          // threads per block (16 wave32)
#define CHUNK 2048         // positions per double-buffer chunk
#define NCH   (SS / CHUNK) // 8

// SCALE = -1/((1-0.5)*ln(0.5)) = 2.8853900817779268
#define JS_SCALE 2.8853900817779268f

// ---------------------------------------------------------------------------
// Per-position JS term.
//   KL1 = ln2 - log1p(sm_l)
//   KL2 = sum_c sm_c * (max(l_c, lnEPS) - logm_c)
//     logm_c (c != label) = clamp(l_c - ln2, lnEPS, 0)   [== log(clip(0.5*sm))]
//     logm_c (c == label) = log1p(sm_l) - ln2            [in range, no clip]
// Equivalent to the reference since log is monotone: log(clip(x,EPS,1)) ==
// clamp(log x, lnEPS, 0) and log(max(x,EPS)) == max(log x, lnEPS).
// ---------------------------------------------------------------------------
__device__ __forceinline__ float jsw_pos(float p0, float p1, float p2, int lab) {
    const float LN2  = 0.6931471805599453f;
    const float LEPS = -16.11809565095832f;   // ln(1e-7)

    float m  = fmaxf(p0, fmaxf(p1, p2));
    float a0 = p0 - m, a1 = p1 - m, a2 = p2 - m;
    float e0 = __expf(a0), e1 = __expf(a1), e2 = __expf(a2);
    float Z  = e0 + e1 + e2;                  // in [1,3]
    float rZ = __builtin_amdgcn_rcpf(Z);
    float lZ = __logf(Z);
    float l0 = a0 - lZ, l1 = a1 - lZ, l2 = a2 - lZ;   // log-softmax
    float s0 = e0 * rZ, s1 = e1 * rZ, s2 = e2 * rZ;   // softmax

    float sl = (lab == 0) ? s0 : ((lab == 1) ? s1 : s2);
    float ll = (lab == 0) ? l0 : ((lab == 1) ? l1 : l2);

    float log1psl = __logf(1.0f + sl);
    float kl1 = LN2 - log1psl;

    float t0 = s0 * (fmaxf(l0, LEPS) - fminf(fmaxf(l0 - LN2, LEPS), 0.0f));
    float t1 = s1 * (fmaxf(l1, LEPS) - fminf(fmaxf(l1 - LN2, LEPS), 0.0f));
    float t2 = s2 * (fmaxf(l2, LEPS) - fminf(fmaxf(l2 - LN2, LEPS), 0.0f));
    float tgen = sl * (fmaxf(ll, LEPS) - fminf(fmaxf(ll - LN2, LEPS), 0.0f));
    float tlab = sl * (fmaxf(ll, LEPS) - (log1psl - LN2));
    float kl2 = t0 + t1 + t2 - tgen + tlab;

    return 0.5f * (kl1 + kl2);
}

// ---------------------------------------------------------------------------
// One block per row b:
//   phase 0: async-prefetch pred chunks 0,1 into LDS (overlaps phase 1)
//   phase 1: find j = first index with label==3 (else S-1)
//   phase 2: double-buffered async pipeline over active chunks; masked sum
//   output : row_out[b] = rowsum / max(j,1)
// ---------------------------------------------------------------------------
__launch_bounds__(NT)
__global__ void js_row_kernel(const float* __restrict__ pred,
                              const int*   __restrict__ labels,
                              float*       __restrict__ row_out) {
    __shared__ __align__(16) float s_pred[2][3][CHUNK];  // 48 KB
    __shared__ int   s_ired[NT];
    __shared__ float s_fred[NT];
    __shared__ int   s_jv;

    const int b   = blockIdx.x;
    const int tid = threadIdx.x;
    const float* predRow = pred + (size_t)b * 4 * SS;     // classes 0..2 used
    const int*   labRow  = labels + (size_t)b * SS;
    const int    e0 = tid * 4;                            // 4 floats / thread / chunk

    // phase 0: prefetch chunks 0 and 1 (3 issues per chunk per wave, in order)
#pragma unroll
    for (int c = 0; c < 3; ++c)
        async_copy_b128(predRow + (size_t)c * SS + e0, &s_pred[0][c][e0]);
#pragma unroll
    for (int c = 0; c < 3; ++c)
        async_copy_b128(predRow + (size_t)c * SS + CHUNK + e0, &s_pred[1][c][e0]);

    // phase 1: first label==3 (min index), overlapped with the async DMA
    const int4* lab4 = reinterpret_cast<const int4*>(labRow);
    int mn = 0x7FFFFFFF;
#pragma unroll
    for (int it = 0; it < SS / (NT * 4); ++it) {          // 8 iterations
        int idx = it * NT + tid;
        int4 lv = lab4[idx];
        int s = idx * 4;
        if (lv.x == 3) mn = min(mn, s + 0);
        if (lv.y == 3) mn = min(mn, s + 1);
        if (lv.z == 3) mn = min(mn, s + 2);
        if (lv.w == 3) mn = min(mn, s + 3);
    }
    s_ired[tid] = mn;
    __syncthreads();
    for (int off = NT / 2; off > 0; off >>= 1) {
        if (tid < off) s_ired[tid] = min(s_ired[tid], s_ired[tid + off]);
        __syncthreads();
    }
    if (tid == 0) s_jv = (s_ired[0] == 0x7FFFFFFF) ? (SS - 1) : s_ired[0];
    __syncthreads();
    const int j = s_jv;
    int nact = (j + CHUNK - 1) / CHUNK;                   // data guarantees >= 4
    if (nact < 2) nact = 2;                               // chunks 0,1 already in flight
    if (nact > NCH) nact = NCH;

    // phase 2: double-buffered compute
    float acc = 0.0f;
    for (int ch = 0; ch < nact; ++ch) {
        if (ch + 1 < nact) wait_asynccnt<3>();            // oldest chunk landed (in-order)
        else               wait_asynccnt<0>();
        __syncthreads();                                  // all waves' portions visible

        const int buf = ch & 1;
        const int sb  = ch * CHUNK + e0;
        if (sb < j) {
            float4 P0 = *reinterpret_cast<const float4*>(&s_pred[buf][0][e0]);
            float4 P1 = *reinterpret_cast<const float4*>(&s_pred[buf][1][e0]);
            float4 P2 = *reinterpret_cast<const float4*>(&s_pred[buf][2][e0]);
            int4   lv = lab4[ch * (CHUNK / 4) + tid];     // L2-hot second read
            float v;
            v = jsw_pos(P0.x, P1.x, P2.x, lv.x); if (sb + 0 < j) acc += v;
            v = jsw_pos(P0.y, P1.y, P2.y, lv.y); if (sb + 1 < j) acc += v;
            v = jsw_pos(P0.z, P1.z, P2.z, lv.z); if (sb + 2 < j) acc += v;
            v = jsw_pos(P0.w, P1.w, P2.w, lv.w); if (sb + 3 < j) acc += v;
        }
        __syncthreads();                                  // everyone done with buf
        if (ch + 2 < nact) {
#pragma unroll
            for (int c = 0; c < 3; ++c)
                async_copy_b128(predRow + (size_t)c * SS + (size_t)(ch + 2) * CHUNK + e0,
                                &s_pred[buf][c][e0]);
        }
    }

    // deterministic block reduction
    s_fred[tid] = acc;
    __syncthreads();
    for (int off = NT / 2; off > 0; off >>= 1) {
        if (tid < off) s_fred[tid] += s_fred[tid + off];
        __syncthreads();
    }
    if (tid == 0) row_out[b] = s_fred[0] / fmaxf((float)j, 1.0f);
}

// ---------------------------------------------------------------------------
// Final deterministic 512-way reduction: out = SCALE * mean_b(row)
// ---------------------------------------------------------------------------
__launch_bounds__(BB)
__global__ void js_final_kernel(const float* __restrict__ row_vals,
                                float* __restrict__ out) {
    __shared__ float s_red[BB];
    const int tid = threadIdx.x;
    s_red[tid] = row_vals[tid];
    __syncthreads();
    for (int off = BB / 2; off > 0; off >>= 1) {
        if (tid < off) s_red[tid] += s_red[tid + off];
        __syncthreads();
    }
    if (tid == 0) out[0] = s_red[0] * (JS_SCALE / (float)BB);
}

// ---------------------------------------------------------------------------
extern "C" void kernel_launch(void* const* d_in, const int* in_sizes, int n_in,
                              void* d_out, int out_size, void* d_ws, size_t ws_size,
                              hipStream_t stream) {
    (void)in_sizes; (void)n_in; (void)out_size; (void)ws_size;
    const float* pred   = (const float*)d_in[0];
    const int*   labels = (const int*)d_in[1];
    float*       rowws  = (float*)d_ws;     // 512 floats
    float*       out    = (float*)d_out;    // 1 float

    js_row_kernel<<<BB, NT, 0, stream>>>(pred, labels, rowws);
    js_final_kernel<<<1, BB, 0, stream>>>(rowws, out);
}